// RNNNameClassifier_69999376990386
// MI455X (gfx1250) — compile-verified
//
#include <hip/hip_runtime.h>
#include <hip/hip_bf16.h>
#include <math.h>

#define INPUT_SZ 128
#define HIDDEN   256
#define OUTSZ    64

typedef float v2f __attribute__((ext_vector_type(2)));
typedef float v4f __attribute__((ext_vector_type(4)));
typedef float v8f __attribute__((ext_vector_type(8)));

// ---------------------------------------------------------------------------
// Kernel 1: x_proj[s][j] = sum_k name[s][k] * W_ih[j][k] + b_ih[j]
// One wave per 16x16 output tile, K accumulated with V_WMMA_F32_16X16X4_F32
// (fp32 kept end-to-end: the recurrence consumes x_proj 262144 times).
// A layout (ISA 7.12.2, 32-bit 16x4): lane<16 -> (K0,K1), lane>=16 -> (K2,K3)
// B layout mirrors A; both become a single v2f (global_load_b64) per lane.
// Verified lowering: chained v_wmma_f32_16x16x4_f32 with s_clause'd b64 loads.
// ---------------------------------------------------------------------------
__global__ __launch_bounds__(128) void xproj_wmma_kernel(
    const float* __restrict__ name, const float* __restrict__ Wih,
    const float* __restrict__ bih, float* __restrict__ xp) {
  const int lane = threadIdx.x;        // 0..31, one wave per threadIdx.y
  const int half = lane >> 4;          // which K pair this half-wave holds
  const int l    = lane & 15;          // tile row (A) / tile col (B, D)
  const int s0   = blockIdx.x * 16;
  const int j0   = (blockIdx.y * 4 + threadIdx.y) * 16;

  const float* arow = name + (size_t)(s0 + l) * INPUT_SZ + 2 * half;
  const float* brow = Wih  + (size_t)(j0 + l) * INPUT_SZ + 2 * half;

  v8f c = {0.f, 0.f, 0.f, 0.f, 0.f, 0.f, 0.f, 0.f};
#pragma unroll
  for (int k0 = 0; k0 < INPUT_SZ; k0 += 4) {
    v2f a = *(const v2f*)(arow + k0);
    v2f b = *(const v2f*)(brow + k0);
    // (neg_a, A, neg_b, B, c_mod, C, reuse_a, reuse_b)
    c = __builtin_amdgcn_wmma_f32_16x16x4_f32(false, a, false, b, (short)0, c,
                                              false, false);
  }
  const float bias = bih[j0 + l];
#pragma unroll
  for (int r = 0; r < 8; ++r) {
    const int row = s0 + r + 8 * half;   // D: VGPR r = row r / r+8 per half
    xp[(size_t)row * HIDDEN + j0 + l] = c[r] + bias;
  }
}

// ---------------------------------------------------------------------------
// Kernel 2: the serial tanh recurrence + final projection + log_softmax.
// Single workgroup of 256 threads (8 wave32 on one WGP).
// Thread j owns output row j: W_hh[j][0..255] lives in 64 v4f VGPR quads
// (verified spill-free: vmem count matches algorithmic traffic exactly).
// Per step: 64 ds_load_b128 broadcast reads of h + 256 contracted FMAs with
// 4 v4f accumulators (16 independent chains, depth 16) -- sits on the WGP's
// f32-FMA issue floor, which probing showed is the best available (no
// V_PK_FMA_F32 / VOPD formation on this toolchain; forcing via asm spills).
// h double-buffered in LDS, one s_barrier per step, NT prefetch of next
// step's x_proj element issued before the FMA block.
// FUSED variant (if d_ws too small for x_proj) also caches W_ih[j][*] in
// registers and streams name rows through LDS.
// ---------------------------------------------------------------------------
template <bool FUSED>
__global__ __launch_bounds__(HIDDEN, 1) void rnn_scan_kernel(
    const float* __restrict__ xp, const float* __restrict__ name,
    const float* __restrict__ Wih, const float* __restrict__ Whh,
    const float* __restrict__ bih, const float* __restrict__ bhh,
    const float* __restrict__ Wout, const float* __restrict__ bout,
    float* __restrict__ out, int S) {
  const int j = threadIdx.x;  // 0..255

  __shared__ __align__(16) float hbuf[2][HIDDEN];
  __shared__ __align__(16) float nbuf[2][INPUT_SZ];
  __shared__ float lred[OUTSZ];
  __shared__ float lse;

  // Register-resident recurrent weights: row j of W_hh as v4f quads.
  v4f w4[HIDDEN / 4];
#pragma unroll
  for (int k = 0; k < HIDDEN; k += 4)
    w4[k / 4] = *(const v4f*)(Whh + (size_t)j * HIDDEN + k);

  v4f wi4[INPUT_SZ / 4];  // only live in the FUSED variant
  if (FUSED) {
#pragma unroll
    for (int k = 0; k < INPUT_SZ; k += 4)
      wi4[k / 4] = *(const v4f*)(Wih + (size_t)j * INPUT_SZ + k);
  }

  const float bj = bhh[j] + (FUSED ? bih[j] : 0.0f);
  hbuf[0][j] = 0.0f;  // h0 = zeros
  if (FUSED && j < INPUT_SZ) nbuf[0][j] = name[j];
  __syncthreads();

  float xp_cur = 0.0f;
  if (!FUSED) xp_cur = __builtin_nontemporal_load(xp + j);

  for (int s = 0; s < S; ++s) {
    const int p = s & 1;

    // Prefetch next step's streaming input before the FMA block (latency hide).
    float xp_nxt = 0.0f;
    float nm_nxt = 0.0f;
    if (!FUSED) {
      if (s + 1 < S)
        xp_nxt = __builtin_nontemporal_load(xp + (size_t)(s + 1) * HIDDEN + j);
    } else {
      if (j < INPUT_SZ && s + 1 < S)
        nm_nxt = __builtin_nontemporal_load(name + (size_t)(s + 1) * INPUT_SZ + j);
    }

    const float* hc = hbuf[p];
    v4f acc0 = {0.f, 0.f, 0.f, 0.f};
    v4f acc1 = {0.f, 0.f, 0.f, 0.f};
    v4f acc2 = {0.f, 0.f, 0.f, 0.f};
    v4f acc3 = {0.f, 0.f, 0.f, 0.f};
#pragma unroll
    for (int k = 0; k < HIDDEN; k += 16) {
      acc0 += w4[k / 4 + 0] * *(const v4f*)(hc + k + 0);
      acc1 += w4[k / 4 + 1] * *(const v4f*)(hc + k + 4);
      acc2 += w4[k / 4 + 2] * *(const v4f*)(hc + k + 8);
      acc3 += w4[k / 4 + 3] * *(const v4f*)(hc + k + 12);
    }
    if (FUSED) {
      const float* nc = nbuf[p];
#pragma unroll
      for (int k = 0; k < INPUT_SZ; k += 16) {
        acc0 += wi4[k / 4 + 0] * *(const v4f*)(nc + k + 0);
        acc1 += wi4[k / 4 + 1] * *(const v4f*)(nc + k + 4);
        acc2 += wi4[k / 4 + 2] * *(const v4f*)(nc + k + 8);
        acc3 += wi4[k / 4 + 3] * *(const v4f*)(nc + k + 12);
      }
    }
    const v4f av = (acc0 + acc1) + (acc2 + acc3);
    const float dot = (av.x + av.y) + (av.z + av.w);
    const float hn = tanhf(xp_cur + bj + dot);

    hbuf[p ^ 1][j] = hn;
    if (FUSED && j < INPUT_SZ) nbuf[p ^ 1][j] = nm_nxt;
    __syncthreads();
    xp_cur = xp_nxt;
  }

  // Epilogue: logits = h @ W_out.T + b_out ; out = log_softmax(logits)
  const float* hf = hbuf[S & 1];
  if (j < OUTSZ) {
    v4f acc = {0.f, 0.f, 0.f, 0.f};
#pragma unroll 4
    for (int k = 0; k < HIDDEN; k += 4)
      acc += *(const v4f*)(Wout + (size_t)j * HIDDEN + k) * *(const v4f*)(hf + k);
    lred[j] = bout[j] + (acc.x + acc.y) + (acc.z + acc.w);
  }
  __syncthreads();
  if (j == 0) {
    float m = lred[0];
    for (int i = 1; i < OUTSZ; ++i) m = fmaxf(m, lred[i]);
    float ssum = 0.f;
    for (int i = 0; i < OUTSZ; ++i) ssum += expf(lred[i] - m);
    lse = m + logf(ssum);
  }
  __syncthreads();
  if (j < OUTSZ) out[j] = lred[j] - lse;
}

// ---------------------------------------------------------------------------
extern "C" void kernel_launch(void* const* d_in, const int* in_sizes, int n_in,
                              void* d_out, int out_size, void* d_ws,
                              size_t ws_size, hipStream_t stream) {
  const float* name = (const float*)d_in[0];
  const float* Wih  = (const float*)d_in[1];
  const float* Whh  = (const float*)d_in[2];
  const float* bih  = (const float*)d_in[3];
  const float* bhh  = (const float*)d_in[4];
  const float* Wout = (const float*)d_in[5];
  const float* bout = (const float*)d_in[6];
  float* outf = (float*)d_out;

  const int S = in_sizes[0] / INPUT_SZ;                    // 262144
  const size_t need = (size_t)S * HIDDEN * sizeof(float);  // 268 MB x_proj

  if (ws_size >= need && (S % 16) == 0) {
    float* xpw = (float*)d_ws;
    dim3 blk(32, 4, 1);                 // 4 waves: 4 N-tiles per block
    dim3 grd(S / 16, HIDDEN / 64, 1);   // 16384 x 4 blocks
    hipLaunchKernelGGL(xproj_wmma_kernel, grd, blk, 0, stream,
                       name, Wih, bih, xpw);
    hipLaunchKernelGGL((rnn_scan_kernel<false>), dim3(1), dim3(HIDDEN), 0,
                       stream, xpw, name, Wih, Whh, bih, bhh, Wout, bout,
                       outf, S);
  } else {
    hipLaunchKernelGGL((rnn_scan_kernel<true>), dim3(1), dim3(HIDDEN), 0,
                       stream, (const float*)nullptr, name, Wih, Whh, bih, bhh,
                       Wout, bout, outf, S);
  }
}